// GCN_14302241095713
// MI455X (gfx1250) — compile-verified
//
#include <hip/hip_runtime.h>
#include <hip/hip_bf16.h>

typedef __attribute__((ext_vector_type(2))) float v2f;
typedef __attribute__((ext_vector_type(8))) float v8f;

#define DF 64          // feature width (D_IN == D_HID == 64)
#define NGRAPH 128
#define DOUT 10

// ---------------- elementwise / setup kernels ----------------

__global__ __launch_bounds__(256) void k_fill(float* __restrict__ p, float v, int n) {
    int t = blockIdx.x * blockDim.x + threadIdx.x;
    if (t < n) p[t] = v;
}

// deg[col[e]] += 1  (self-loop "1" comes from the k_fill init)
__global__ __launch_bounds__(256) void k_deg(const int* __restrict__ col, float* __restrict__ deg, int E) {
    int t = blockIdx.x * blockDim.x + threadIdx.x;
    if (t < E) atomicAdd(&deg[col[t]], 1.0f);
}

__global__ __launch_bounds__(256) void k_rsqrt(float* __restrict__ d, int n) {
    int t = blockIdx.x * blockDim.x + threadIdx.x;
    if (t < n) d[t] = rsqrtf(d[t]);     // deg >= 1 always (self loop)
}

// norm[e] = dinv[row[e]] * dinv[col[e]]
__global__ __launch_bounds__(256) void k_norm(const int* __restrict__ edge,
                                              const float* __restrict__ dinv,
                                              float* __restrict__ normv, int E) {
    int t = blockIdx.x * blockDim.x + threadIdx.x;
    if (t < E) normv[t] = dinv[edge[t]] * dinv[edge[E + t]];
}

// ---------------- WMMA GEMM: C[nrows x 64] = A[nrows x 64] * W[64 x 64] ----------------
// One wave32 computes one 16-row x 64-col output tile using V_WMMA_F32_16X16X4_F32.
// A fragment (16x4, MxK):  lane 0-15 -> M=lane,   VGPR0:K=0, VGPR1:K=1
//                          lane16-31 -> M=lane-16,VGPR0:K=2, VGPR1:K=3
// B fragment (4x16, KxN):  VGPR r: lanes0-15 K=r (N=lane), lanes16-31 K=r+2 (N=lane-16)
// C/D (16x16):             VGPR r: lanes0-15 M=r,          lanes16-31 M=r+8
__global__ __launch_bounds__(256) void k_gemm64_wmma(const float* __restrict__ A,
                                                     const float* __restrict__ W,
                                                     float* __restrict__ C, int nrows) {
    int wave = blockIdx.x * (blockDim.x >> 5) + (threadIdx.x >> 5);
    int lane = threadIdx.x & 31;
    int m0 = wave * 16;
    if (m0 >= nrows) return;                  // wave-uniform exit: live waves keep EXEC all-1s

    int laneM  = lane & 15;
    int kHalf  = lane >> 4;                   // 0 or 1
    const float* arow = A + (size_t)(m0 + laneM) * DF;

    v8f acc0 = {}, acc1 = {}, acc2 = {}, acc3 = {};

    #pragma unroll
    for (int k0 = 0; k0 < DF; k0 += 4) {
        int ka = k0 + 2 * kHalf;
        v2f a;
        a.x = arow[ka];
        a.y = arow[ka + 1];

        const float* wp = W + (size_t)ka * DF + laneM;
        v2f b0, b1, b2, b3;
        b0.x = wp[0];       b0.y = wp[DF];
        b1.x = wp[16];      b1.y = wp[DF + 16];
        b2.x = wp[32];      b2.y = wp[DF + 32];
        b3.x = wp[48];      b3.y = wp[DF + 48];

        acc0 = __builtin_amdgcn_wmma_f32_16x16x4_f32(false, a, false, b0, (short)0, acc0, false, false);
        acc1 = __builtin_amdgcn_wmma_f32_16x16x4_f32(false, a, false, b1, (short)0, acc1, false, false);
        acc2 = __builtin_amdgcn_wmma_f32_16x16x4_f32(false, a, false, b2, (short)0, acc2, false, false);
        acc3 = __builtin_amdgcn_wmma_f32_16x16x4_f32(false, a, false, b3, (short)0, acc3, false, false);
    }

    // store D: row = r + 8*kHalf, col = nb*16 + laneM
    float* crow = C + (size_t)(m0 + 8 * kHalf) * DF + laneM;
    #pragma unroll
    for (int r = 0; r < 8; ++r) {
        crow[(size_t)r * DF +  0] = acc0[r];
        crow[(size_t)r * DF + 16] = acc1[r];
        crow[(size_t)r * DF + 32] = acc2[r];
        crow[(size_t)r * DF + 48] = acc3[r];
    }
}

// ---------------- aggregation ----------------

// self-loop term also initializes the accumulator: hacc[i][f] = hlin[i][f] * dinv[i]^2
__global__ __launch_bounds__(256) void k_selfinit(const float* __restrict__ hlin,
                                                  const float* __restrict__ dinv,
                                                  float* __restrict__ hacc, int total) {
    int t = blockIdx.x * blockDim.x + threadIdx.x;
    if (t < total) {
        int i = t >> 6;
        float di = dinv[i];
        hacc[t] = hlin[t] * di * di;
    }
}

// per (edge, feature): hacc[col][f] += hlin[row][f] * norm[e]
__global__ __launch_bounds__(256) void k_scatter(const int* __restrict__ edge,
                                                 const float* __restrict__ normv,
                                                 const float* __restrict__ hlin,
                                                 float* __restrict__ hacc, int E) {
    int t = blockIdx.x * blockDim.x + threadIdx.x;
    int e = t >> 6;
    if (e < E) {
        int f = t & 63;
        int r = edge[e];
        int c = edge[E + e];
        float v = hlin[(size_t)r * DF + f] * normv[e];
        atomicAdd(&hacc[(size_t)c * DF + f], v);
    }
}

__global__ __launch_bounds__(256) void k_bias_relu(const float* __restrict__ hacc,
                                                   const float* __restrict__ b,
                                                   float* __restrict__ h, int total) {
    int t = blockIdx.x * blockDim.x + threadIdx.x;
    if (t < total) h[t] = fmaxf(hacc[t] + b[t & 63], 0.0f);
}

// ---------------- pooling + FC ----------------

__global__ __launch_bounds__(256) void k_pool(const float* __restrict__ h,
                                              const int* __restrict__ batch,
                                              float* __restrict__ pooled,
                                              float* __restrict__ cnt, int total) {
    int t = blockIdx.x * blockDim.x + threadIdx.x;
    if (t < total) {
        int i = t >> 6;
        int f = t & 63;
        int g = batch[i];
        atomicAdd(&pooled[(size_t)g * DF + f], h[t]);
        if (f == 0) atomicAdd(&cnt[g], 1.0f);
    }
}

__global__ __launch_bounds__(256) void k_fc(const float* __restrict__ pooled,
                                            const float* __restrict__ cnt,
                                            const float* __restrict__ Wfc,
                                            const float* __restrict__ bfc,
                                            float* __restrict__ out) {
    int t = blockIdx.x * blockDim.x + threadIdx.x;
    if (t < NGRAPH * DOUT) {
        int g = t / DOUT;
        int o = t % DOUT;
        float inv = 1.0f / fmaxf(cnt[g], 1.0f);
        float s = 0.0f;
        #pragma unroll
        for (int k = 0; k < DF; ++k)
            s += pooled[(size_t)g * DF + k] * Wfc[(size_t)k * DOUT + o];
        out[t] = s * inv + bfc[o];
    }
}

// ---------------- launcher ----------------

extern "C" void kernel_launch(void* const* d_in, const int* in_sizes, int n_in,
                              void* d_out, int out_size, void* d_ws, size_t ws_size,
                              hipStream_t stream) {
    const float* x    = (const float*)d_in[0];
    const float* W1   = (const float*)d_in[1];
    const float* b1   = (const float*)d_in[2];
    const float* W2   = (const float*)d_in[3];
    const float* b2   = (const float*)d_in[4];
    const float* Wfc  = (const float*)d_in[5];
    const float* bfc  = (const float*)d_in[6];
    const int*   edge = (const int*)d_in[7];   // [2, E] flat: row = edge[0..E), col = edge[E..2E)
    const int*   batch= (const int*)d_in[8];

    const int N = in_sizes[0] / DF;            // 100000
    const int E = in_sizes[7] / 2;             // 1200000
    float* out = (float*)d_out;

    // workspace carve-up
    float* dinv   = (float*)d_ws;                          // N
    float* normv  = dinv  + ((N + 63) & ~63);              // E
    float* buf0   = normv + ((E + 63) & ~63);              // N*64
    float* buf1   = buf0  + (size_t)N * DF;                // N*64
    float* pooled = buf1  + (size_t)N * DF;                // 128*64
    float* cnt    = pooled + NGRAPH * DF;                  // 128

    const int TB = 256;
    const int totalNF   = N * DF;
    const int gridNF    = (totalNF + TB - 1) / TB;
    const int gridN     = (N + TB - 1) / TB;
    const int gridE     = (E + TB - 1) / TB;
    const int gridEF    = (E * DF + TB - 1) / TB;          // 76.8M threads / 256
    const int gemmBlocks = ((N / 16) + 7) / 8;             // 8 waves/block, 1 tile/wave

    // degrees & normalization
    k_fill  <<<gridN, TB, 0, stream>>>(dinv, 1.0f, N);                 // self-loop contribution
    k_deg   <<<gridE, TB, 0, stream>>>(edge + E, dinv, E);
    k_rsqrt <<<gridN, TB, 0, stream>>>(dinv, N);
    k_norm  <<<gridE, TB, 0, stream>>>(edge, dinv, normv, E);

    // ---- layer 1: buf0 = x@W1 ; buf1 = D^-1/2 A D^-1/2 buf0 ; buf0 = relu(buf1+b1)
    k_gemm64_wmma <<<gemmBlocks, TB, 0, stream>>>(x, W1, buf0, N);
    k_selfinit    <<<gridNF, TB, 0, stream>>>(buf0, dinv, buf1, totalNF);
    k_scatter     <<<gridEF, TB, 0, stream>>>(edge, normv, buf0, buf1, E);
    k_bias_relu   <<<gridNF, TB, 0, stream>>>(buf1, b1, buf0, totalNF);

    // ---- layer 2: buf1 = buf0@W2 ; buf0 = aggregate(buf1) ; buf1 = relu(buf0+b2)
    k_gemm64_wmma <<<gemmBlocks, TB, 0, stream>>>(buf0, W2, buf1, N);
    k_selfinit    <<<gridNF, TB, 0, stream>>>(buf1, dinv, buf0, totalNF);
    k_scatter     <<<gridEF, TB, 0, stream>>>(edge, normv, buf1, buf0, E);
    k_bias_relu   <<<gridNF, TB, 0, stream>>>(buf0, b2, buf1, totalNF);

    // ---- mean pool + FC
    k_fill <<<(NGRAPH * DF + NGRAPH + TB - 1) / TB, TB, 0, stream>>>(pooled, 0.0f, NGRAPH * DF + NGRAPH);
    k_pool <<<gridNF, TB, 0, stream>>>(buf1, batch, pooled, cnt, totalNF);
    k_fc   <<<(NGRAPH * DOUT + TB - 1) / TB, TB, 0, stream>>>(pooled, cnt, Wfc, bfc, out);
}